// TSP_Decoder_20718922235921
// MI455X (gfx1250) — compile-verified
//
#include <hip/hip_runtime.h>

typedef float v2f __attribute__((ext_vector_type(2)));
typedef float v8f __attribute__((ext_vector_type(8)));

#define B_    64
#define POMO_ 200
#define N_    1000
#define EMB_  128
#define H_    8
#define D_    16

#define NT_   63   // ceil(1000/16)
#define PT_   13   // ceil(200/16)

__device__ __forceinline__ v8f wmma4(v2f a, v2f b, v8f c) {
    // V_WMMA_F32_16X16X4_F32 : D = A(16x4) * B(4x16) + C(16x16), all f32
    return __builtin_amdgcn_wmma_f32_16x16x4_f32(
        /*neg_a=*/false, a, /*neg_b=*/false, b,
        /*c_mod=*/(short)0, c, /*reuse_a=*/false, /*reuse_b=*/false);
}

// ---------------------------------------------------------------------------
// Kernel 1: K = enc @ Wk^T, V = enc @ Wv^T   (stored [B, N, H*D] row-major)
// ---------------------------------------------------------------------------
__global__ __launch_bounds__(256) void kv_kernel(
    const float* __restrict__ enc, const float* __restrict__ Wk,
    const float* __restrict__ Wv, float* __restrict__ Kb, float* __restrict__ Vb) {
    __shared__ float X[16 * EMB_];
    const int tileN = blockIdx.x;           // 0..62
    const int b     = blockIdx.y;
    const int node0 = tileN * 16;
    const int tid   = threadIdx.x;

    for (int i = tid; i < (16 * EMB_) / 4; i += 256) {
        int row = (i * 4) >> 7, col = (i * 4) & 127;
        int nr = node0 + row; if (nr >= N_) nr = N_ - 1;
        *reinterpret_cast<float4*>(X + row * EMB_ + col) =
            *reinterpret_cast<const float4*>(enc + ((size_t)b * N_ + nr) * EMB_ + col);
    }
    __syncthreads();

    const int wave = tid >> 5, lane = tid & 31;
    const int lo = lane & 15, hi = lane >> 4;
    const int colb = wave * 16;

    v8f ck = {}, cv = {};
    const float* wkrow = Wk + (size_t)(colb + lo) * EMB_;
    const float* wvrow = Wv + (size_t)(colb + lo) * EMB_;
#pragma unroll 8
    for (int kk = 0; kk < EMB_; kk += 4) {
        v2f a  = *reinterpret_cast<const v2f*>(X + lo * EMB_ + kk + 2 * hi);
        v2f bk = *reinterpret_cast<const v2f*>(wkrow + kk + 2 * hi);
        v2f bv = *reinterpret_cast<const v2f*>(wvrow + kk + 2 * hi);
        ck = wmma4(a, bk, ck);
        cv = wmma4(a, bv, cv);
    }
#pragma unroll
    for (int r = 0; r < 8; ++r) {
        int m = node0 + r + 8 * hi;
        if (m < N_) {
            size_t o = ((size_t)b * N_ + m) * EMB_ + colb + lo;
            Kb[o] = ck[r];
            Vb[o] = cv[r];
        }
    }
}

// ---------------------------------------------------------------------------
// Kernel 2: Q = q1 @ Wqf^T + last @ Wql^T + (last @ lqA^T) @ lqB^T
// ---------------------------------------------------------------------------
__global__ __launch_bounds__(256) void q_kernel(
    const float* __restrict__ q1e, const float* __restrict__ laste,
    const float* __restrict__ Wqf, const float* __restrict__ Wql,
    const float* __restrict__ lqA, const float* __restrict__ lqB,
    float* __restrict__ Qb) {
    __shared__ float X1[16 * EMB_], X2[16 * EMB_], T[16 * 16];
    const int tileP = blockIdx.x, b = blockIdx.y;
    const int p0 = tileP * 16, tid = threadIdx.x;

    for (int i = tid; i < (16 * EMB_) / 4; i += 256) {
        int row = (i * 4) >> 7, col = (i * 4) & 127;
        int pr = p0 + row; if (pr >= POMO_) pr = POMO_ - 1;
        size_t s = ((size_t)b * POMO_ + pr) * EMB_ + col;
        *reinterpret_cast<float4*>(X1 + row * EMB_ + col) = *reinterpret_cast<const float4*>(q1e + s);
        *reinterpret_cast<float4*>(X2 + row * EMB_ + col) = *reinterpret_cast<const float4*>(laste + s);
    }
    __syncthreads();

    const int wave = tid >> 5, lane = tid & 31;
    const int lo = lane & 15, hi = lane >> 4;

    if (wave == 0) {   // 16x16 LoRA intermediate: T = last_tile @ lqA^T
        v8f t = {};
#pragma unroll 8
        for (int kk = 0; kk < EMB_; kk += 4) {
            v2f a  = *reinterpret_cast<const v2f*>(X2 + lo * EMB_ + kk + 2 * hi);
            v2f bb = *reinterpret_cast<const v2f*>(lqA + lo * EMB_ + kk + 2 * hi);
            t = wmma4(a, bb, t);
        }
#pragma unroll
        for (int r = 0; r < 8; ++r) T[(r + 8 * hi) * 16 + lo] = t[r];
    }
    __syncthreads();

    const int colb = wave * 16;
    v8f acc = {};
#pragma unroll 8
    for (int kk = 0; kk < EMB_; kk += 4) {
        v2f a1 = *reinterpret_cast<const v2f*>(X1 + lo * EMB_ + kk + 2 * hi);
        v2f b1 = *reinterpret_cast<const v2f*>(Wqf + (size_t)(colb + lo) * EMB_ + kk + 2 * hi);
        acc = wmma4(a1, b1, acc);
        v2f a2 = *reinterpret_cast<const v2f*>(X2 + lo * EMB_ + kk + 2 * hi);
        v2f b2 = *reinterpret_cast<const v2f*>(Wql + (size_t)(colb + lo) * EMB_ + kk + 2 * hi);
        acc = wmma4(a2, b2, acc);
    }
#pragma unroll
    for (int kk = 0; kk < 16; kk += 4) {   // rank-16 LoRA-B (LORA_SCALE == 1)
        v2f a  = *reinterpret_cast<const v2f*>(T + lo * 16 + kk + 2 * hi);
        v2f bb = *reinterpret_cast<const v2f*>(lqB + (size_t)(colb + lo) * 16 + kk + 2 * hi);
        acc = wmma4(a, bb, acc);
    }
#pragma unroll
    for (int r = 0; r < 8; ++r) {
        int p = p0 + r + 8 * hi;
        if (p < POMO_) Qb[((size_t)b * POMO_ + p) * EMB_ + colb + lo] = acc[r];
    }
}

// ---------------------------------------------------------------------------
// Kernel 3: flash-style masked MHA; one wave per (p-tile, head, batch)
// ---------------------------------------------------------------------------
__global__ __launch_bounds__(32) void attn_kernel(
    const float* __restrict__ Qb, const float* __restrict__ Kb,
    const float* __restrict__ Vb, const float* __restrict__ mask,
    float* __restrict__ Ob) {
    __shared__ float P[16 * 16];
    const int pt = blockIdx.x, h = blockIdx.y, b = blockIdx.z;
    const int p0 = pt * 16;
    const int lane = threadIdx.x, lo = lane & 15, hi = lane >> 4;

    int qrow = p0 + lo; if (qrow >= POMO_) qrow = POMO_ - 1;
    v2f qa[4];
#pragma unroll
    for (int t = 0; t < 4; ++t)
        qa[t] = *reinterpret_cast<const v2f*>(
            Qb + ((size_t)b * POMO_ + qrow) * EMB_ + h * 16 + 4 * t + 2 * hi);

    v8f O = {};
    float mrun[8], srun[8];
#pragma unroll
    for (int r = 0; r < 8; ++r) { mrun[r] = -3.0e38f; srun[r] = 0.0f; }

    for (int nt = 0; nt < NT_; ++nt) {
        const int node0 = nt * 16;
        v8f sc = {};
        int kn = node0 + lo; if (kn >= N_) kn = N_ - 1;
        const float* kbase = Kb + ((size_t)b * N_ + kn) * EMB_ + h * 16;
#pragma unroll
        for (int t = 0; t < 4; ++t) {
            v2f bb = *reinterpret_cast<const v2f*>(kbase + 4 * t + 2 * hi);
            sc = wmma4(qa[t], bb, sc);
        }
        const int n = node0 + lo;
#pragma unroll
        for (int r = 0; r < 8; ++r) {
            int p = p0 + r + 8 * hi;
            int pc = (p < POMO_) ? p : POMO_ - 1;
            float val = (n < N_)
                ? sc[r] * 0.25f + mask[((size_t)b * POMO_ + pc) * N_ + n]
                : -3.0e38f;
            float tm = val;                               // row max over 16 lanes
            tm = fmaxf(tm, __shfl_xor(tm, 1, 32));
            tm = fmaxf(tm, __shfl_xor(tm, 2, 32));
            tm = fmaxf(tm, __shfl_xor(tm, 4, 32));
            tm = fmaxf(tm, __shfl_xor(tm, 8, 32));
            float nm = fmaxf(mrun[r], tm);
            float scale = expf(mrun[r] - nm);
            float pv = expf(val - nm);
            float ps = pv;                                // row sum over 16 lanes
            ps += __shfl_xor(ps, 1, 32);
            ps += __shfl_xor(ps, 2, 32);
            ps += __shfl_xor(ps, 4, 32);
            ps += __shfl_xor(ps, 8, 32);
            srun[r] = srun[r] * scale + ps;
            mrun[r] = nm;
            O[r] *= scale;
            P[(r + 8 * hi) * 16 + lo] = pv;               // C-layout -> LDS
        }
        __syncthreads();                                  // single-wave WG: S_NOP
#pragma unroll
        for (int t = 0; t < 4; ++t) {                     // O += P @ V_tile
            v2f aP = *reinterpret_cast<const v2f*>(P + lo * 16 + 4 * t + 2 * hi);
            int k0 = node0 + 4 * t + 2 * hi, k1 = k0 + 1;
            if (k0 >= N_) k0 = N_ - 1;
            if (k1 >= N_) k1 = N_ - 1;
            v2f bv;
            bv.x = Vb[((size_t)b * N_ + k0) * EMB_ + h * 16 + lo];
            bv.y = Vb[((size_t)b * N_ + k1) * EMB_ + h * 16 + lo];
            O = wmma4(aP, bv, O);
        }
        __syncthreads();
    }
#pragma unroll
    for (int r = 0; r < 8; ++r) {
        int p = p0 + r + 8 * hi;
        if (p < POMO_)
            Ob[((size_t)b * POMO_ + p) * EMB_ + h * 16 + lo] = O[r] / srun[r];
    }
}

// ---------------------------------------------------------------------------
// Kernel 4: mh = outc @ Wcomb^T + bcomb + (outc @ lcA^T) @ lcB^T
// ---------------------------------------------------------------------------
__global__ __launch_bounds__(256) void comb_kernel(
    const float* __restrict__ Oc, const float* __restrict__ Wc,
    const float* __restrict__ bc, const float* __restrict__ lcA,
    const float* __restrict__ lcB, float* __restrict__ Mh) {
    __shared__ float X[16 * EMB_], T[16 * 16];
    const int tileP = blockIdx.x, b = blockIdx.y;
    const int p0 = tileP * 16, tid = threadIdx.x;

    for (int i = tid; i < (16 * EMB_) / 4; i += 256) {
        int row = (i * 4) >> 7, col = (i * 4) & 127;
        int pr = p0 + row; if (pr >= POMO_) pr = POMO_ - 1;
        *reinterpret_cast<float4*>(X + row * EMB_ + col) =
            *reinterpret_cast<const float4*>(Oc + ((size_t)b * POMO_ + pr) * EMB_ + col);
    }
    __syncthreads();

    const int wave = tid >> 5, lane = tid & 31;
    const int lo = lane & 15, hi = lane >> 4;

    if (wave == 0) {
        v8f t = {};
#pragma unroll 8
        for (int kk = 0; kk < EMB_; kk += 4) {
            v2f a  = *reinterpret_cast<const v2f*>(X + lo * EMB_ + kk + 2 * hi);
            v2f bb = *reinterpret_cast<const v2f*>(lcA + lo * EMB_ + kk + 2 * hi);
            t = wmma4(a, bb, t);
        }
#pragma unroll
        for (int r = 0; r < 8; ++r) T[(r + 8 * hi) * 16 + lo] = t[r];
    }
    __syncthreads();

    const int colb = wave * 16;
    const float bias = bc[colb + lo];
    v8f acc = {};
#pragma unroll
    for (int r = 0; r < 8; ++r) acc[r] = bias;
#pragma unroll 8
    for (int kk = 0; kk < EMB_; kk += 4) {
        v2f a  = *reinterpret_cast<const v2f*>(X + lo * EMB_ + kk + 2 * hi);
        v2f bb = *reinterpret_cast<const v2f*>(Wc + (size_t)(colb + lo) * EMB_ + kk + 2 * hi);
        acc = wmma4(a, bb, acc);
    }
#pragma unroll
    for (int kk = 0; kk < 16; kk += 4) {
        v2f a  = *reinterpret_cast<const v2f*>(T + lo * 16 + kk + 2 * hi);
        v2f bb = *reinterpret_cast<const v2f*>(lcB + (size_t)(colb + lo) * 16 + kk + 2 * hi);
        acc = wmma4(a, bb, acc);
    }
#pragma unroll
    for (int r = 0; r < 8; ++r) {
        int p = p0 + r + 8 * hi;
        if (p < POMO_) Mh[((size_t)b * POMO_ + p) * EMB_ + colb + lo] = acc[r];
    }
}

// ---------------------------------------------------------------------------
// Kernel 5: probs = softmax(10*tanh((mh @ enc^T)/sqrt(EMB)) + mask)
// ---------------------------------------------------------------------------
__global__ __launch_bounds__(256) void final_kernel(
    const float* __restrict__ Mh, const float* __restrict__ enc,
    const float* __restrict__ mask, float* __restrict__ out) {
    __shared__ float X[16 * EMB_];
    __shared__ float wmax[8 * 16], wsum[8 * 16], Mrow[16], Srow[16];
    const int tileP = blockIdx.x, b = blockIdx.y;
    const int p0 = tileP * 16, tid = threadIdx.x;

    for (int i = tid; i < (16 * EMB_) / 4; i += 256) {
        int row = (i * 4) >> 7, col = (i * 4) & 127;
        int pr = p0 + row; if (pr >= POMO_) pr = POMO_ - 1;
        *reinterpret_cast<float4*>(X + row * EMB_ + col) =
            *reinterpret_cast<const float4*>(Mh + ((size_t)b * POMO_ + pr) * EMB_ + col);
    }
    __syncthreads();

    const int wave = tid >> 5, lane = tid & 31;
    const int lo = lane & 15, hi = lane >> 4;
    const float inv_se = 1.0f / 11.313708498984761f;

    float mrun[8], srun[8];
#pragma unroll
    for (int r = 0; r < 8; ++r) { mrun[r] = -3.0e38f; srun[r] = 0.0f; }

    for (int nt = wave; nt < NT_; nt += 8) {
        const int node0 = nt * 16;
        v8f sc = {};
        int nr = node0 + lo; if (nr >= N_) nr = N_ - 1;
        const float* ebase = enc + ((size_t)b * N_ + nr) * EMB_;
#pragma unroll 8
        for (int kk = 0; kk < EMB_; kk += 4) {
            v2f a  = *reinterpret_cast<const v2f*>(X + lo * EMB_ + kk + 2 * hi);
            v2f bb = *reinterpret_cast<const v2f*>(ebase + kk + 2 * hi);
            sc = wmma4(a, bb, sc);
        }
        const int n = node0 + lo;
#pragma unroll
        for (int r = 0; r < 8; ++r) {
            int p = p0 + r + 8 * hi;
            int pc = (p < POMO_) ? p : POMO_ - 1;
            float val = (n < N_)
                ? 10.0f * tanhf(sc[r] * inv_se) + mask[((size_t)b * POMO_ + pc) * N_ + n]
                : -3.0e38f;
            if (p < POMO_ && n < N_)
                out[((size_t)b * POMO_ + p) * N_ + n] = val;   // raw score, pass 2 fixes
            float tm = val;
            tm = fmaxf(tm, __shfl_xor(tm, 1, 32));
            tm = fmaxf(tm, __shfl_xor(tm, 2, 32));
            tm = fmaxf(tm, __shfl_xor(tm, 4, 32));
            tm = fmaxf(tm, __shfl_xor(tm, 8, 32));
            float nm = fmaxf(mrun[r], tm);
            float scale = expf(mrun[r] - nm);
            float pv = expf(val - nm);
            float ps = pv;
            ps += __shfl_xor(ps, 1, 32);
            ps += __shfl_xor(ps, 2, 32);
            ps += __shfl_xor(ps, 4, 32);
            ps += __shfl_xor(ps, 8, 32);
            srun[r] = srun[r] * scale + ps;
            mrun[r] = nm;
        }
    }
    if (lo == 0) {
#pragma unroll
        for (int r = 0; r < 8; ++r) {
            wmax[wave * 16 + r + 8 * hi] = mrun[r];
            wsum[wave * 16 + r + 8 * hi] = srun[r];
        }
    }
    __syncthreads();
    if (tid < 16) {
        float M = -3.0e38f;
        for (int w = 0; w < 8; ++w) M = fmaxf(M, wmax[w * 16 + tid]);
        float S = 0.0f;
        for (int w = 0; w < 8; ++w) S += wsum[w * 16 + tid] * expf(wmax[w * 16 + tid] - M);
        Mrow[tid] = M;
        Srow[tid] = (S > 0.0f) ? S : 1.0f;
    }
    __syncthreads();
    for (int nt = wave; nt < NT_; nt += 8) {   // normalize: same wave re-reads its stores
        const int n = nt * 16 + lo;
#pragma unroll
        for (int r = 0; r < 8; ++r) {
            int p = p0 + r + 8 * hi;
            if (p < POMO_ && n < N_) {
                size_t o = ((size_t)b * POMO_ + p) * N_ + n;
                float x = out[o];
                out[o] = expf(x - Mrow[r + 8 * hi]) / Srow[r + 8 * hi];
            }
        }
    }
}

// ---------------------------------------------------------------------------
extern "C" void kernel_launch(void* const* d_in, const int* in_sizes, int n_in,
                              void* d_out, int out_size, void* d_ws, size_t ws_size,
                              hipStream_t stream) {
    (void)in_sizes; (void)n_in; (void)out_size; (void)ws_size;
    const float* enc   = (const float*)d_in[0];
    const float* q1e   = (const float*)d_in[1];
    const float* laste = (const float*)d_in[2];
    const float* mask  = (const float*)d_in[3];
    const float* Wqf   = (const float*)d_in[4];
    const float* Wql   = (const float*)d_in[5];
    const float* Wk    = (const float*)d_in[6];
    const float* Wv    = (const float*)d_in[7];
    const float* Wc    = (const float*)d_in[8];
    const float* bc    = (const float*)d_in[9];
    const float* lqA   = (const float*)d_in[10];
    const float* lqB   = (const float*)d_in[11];
    const float* lcA   = (const float*)d_in[12];
    const float* lcB   = (const float*)d_in[13];
    float* out = (float*)d_out;

    const size_t KV = (size_t)B_ * N_ * EMB_;       // 8,192,000 floats
    const size_t QS = (size_t)B_ * POMO_ * EMB_;    // 1,638,400 floats
    float* Kb  = (float*)d_ws;
    float* Vb  = Kb + KV;
    float* Qb  = Vb + KV;
    float* Ob  = Qb + QS;
    float* Mhb = Ob + QS;

    kv_kernel   <<<dim3(NT_, B_),      256, 0, stream>>>(enc, Wk, Wv, Kb, Vb);
    q_kernel    <<<dim3(PT_, B_),      256, 0, stream>>>(q1e, laste, Wqf, Wql, lqA, lqB, Qb);
    attn_kernel <<<dim3(PT_, H_, B_),   32, 0, stream>>>(Qb, Kb, Vb, mask, Ob);
    comb_kernel <<<dim3(PT_, B_),      256, 0, stream>>>(Ob, Wc, bc, lcA, lcB, Mhb);
    final_kernel<<<dim3(PT_, B_),      256, 0, stream>>>(Mhb, enc, mask, out);
}